// NonLinearNeuralNetwork_3315714752773
// MI455X (gfx1250) — compile-verified
//
#include <hip/hip_runtime.h>

typedef float v2f __attribute__((ext_vector_type(2)));
typedef float v8f __attribute__((ext_vector_type(8)));

// D = A(16x4) * B(4x16) + C(16x16), fp32, one wave.
__device__ __forceinline__ v8f wmma_f32_4(v2f a, v2f b, v8f c) {
    return __builtin_amdgcn_wmma_f32_16x16x4_f32(
        /*neg_a=*/false, a, /*neg_b=*/false, b,
        /*c_mod=*/(short)0, c, /*reuse_a=*/false, /*reuse_b=*/false);
}

template <bool GUARD>
__launch_bounds__(256)
__global__ void mlp3_wmma_kernel(const float* __restrict__ x,
                                 const float* __restrict__ W1, const float* __restrict__ b1,
                                 const float* __restrict__ W2, const float* __restrict__ b2,
                                 const float* __restrict__ W3, const float* __restrict__ b3,
                                 float* __restrict__ out,
                                 int Brows, int tiles)
{
    // one 16x16 fp32 staging tile per wave (8 waves/block -> 8KB)
    __shared__ float lds[8][16 * 16];

    const int tid  = threadIdx.x;
    const int wave = tid >> 5;          // wave32
    const int lane = tid & 31;
    const int nlo  = lane & 15;         // N (or M) index within half-wave
    const int hi   = lane >> 4;         // 0 = lanes 0-15, 1 = lanes 16-31
    float* L = lds[wave];

    // ---------------- weights into WMMA B-layout (once per wave) -------------
    // B layout (4x16, KxN): lane holds N=nlo; K = 2*hi + {0 (vgpr0), 1 (vgpr1)}.
    v2f B1;
    B1.x = W1[(2 * hi + 0) * 16 + nlo];
    B1.y = W1[(2 * hi + 1) * 16 + nlo];

    // Layer 2: W2 is (16,8); pad N 8->16 with zeros. 4 K-steps of 4.
    v2f B2[4];
#pragma unroll
    for (int j = 0; j < 4; ++j) {
        const int k0 = 4 * j + 2 * hi;
        B2[j].x = (nlo < 8) ? W2[(k0 + 0) * 8 + nlo] : 0.0f;
        B2[j].y = (nlo < 8) ? W2[(k0 + 1) * 8 + nlo] : 0.0f;
    }

    // Layer 3: W3 is (8,3); pad N 3->16 with zeros. 2 K-steps of 4.
    v2f B3[2];
#pragma unroll
    for (int j = 0; j < 2; ++j) {
        const int k0 = 4 * j + 2 * hi;
        B3[j].x = (nlo < 3) ? W3[(k0 + 0) * 3 + nlo] : 0.0f;
        B3[j].y = (nlo < 3) ? W3[(k0 + 1) * 3 + nlo] : 0.0f;
    }

    // Bias broadcast values for C tiles (C[m][n] = b[n]).
    const float c1 = b1[nlo];
    const float c2 = (nlo < 8) ? b2[nlo] : 0.0f;
    const float c3 = (nlo < 3) ? b3[nlo] : 0.0f;

    // LDS addressing: store D-layout tile row-major, load in A-layout.
    const int st_base = hi * 128 + nlo;       // row = vgpr_i + 8*hi, col = nlo
    const int ld_base = nlo * 16 + hi * 2;    // row = nlo, col = 4*j + 2*hi + {0,1}

    const int wave_stride = gridDim.x * 8;    // 8 waves per block

    for (int t = blockIdx.x * 8 + wave; t < tiles; t += wave_stride) {
        const int base = t * 16;

        // ---- load A1 directly in WMMA A-layout: lane -> x[base+nlo][2*hi..+1]
        const int row = base + nlo;
        v2f a1;
        if (GUARD) {
            a1 = (v2f){0.0f, 0.0f};
            if (row < Brows)
                a1 = __builtin_nontemporal_load((const v2f*)(x + row * 4 + 2 * hi));
        } else {
            a1 = __builtin_nontemporal_load((const v2f*)(x + row * 4 + 2 * hi));
        }

        // ---- layer 1: h1 = relu(x @ W1 + b1), one WMMA (K=4 exact) ----------
        v8f acc;
#pragma unroll
        for (int i = 0; i < 8; ++i) acc[i] = c1;
        v8f h1 = wmma_f32_4(a1, B1, acc);
#pragma unroll
        for (int i = 0; i < 8; ++i) h1[i] = fmaxf(h1[i], 0.0f);

        // ---- transpose D-layout -> A-layout through wave-private LDS --------
#pragma unroll
        for (int i = 0; i < 8; ++i) L[st_base + i * 16] = h1[i];
        __asm__ volatile("s_wait_dscnt 0x0" ::: "memory");
        v2f a2[4];
#pragma unroll
        for (int j = 0; j < 4; ++j) a2[j] = *(const v2f*)(L + ld_base + 4 * j);

        // ---- layer 2: h2 = relu(h1 @ W2 + b2), K=16 via 4 chained WMMAs -----
#pragma unroll
        for (int i = 0; i < 8; ++i) acc[i] = c2;
        v8f h2 = wmma_f32_4(a2[0], B2[0], acc);
        h2 = wmma_f32_4(a2[1], B2[1], h2);
        h2 = wmma_f32_4(a2[2], B2[2], h2);
        h2 = wmma_f32_4(a2[3], B2[3], h2);
#pragma unroll
        for (int i = 0; i < 8; ++i) h2[i] = fmaxf(h2[i], 0.0f);

        // ---- second LDS round-trip (only cols 0..7 are consumed) ------------
#pragma unroll
        for (int i = 0; i < 8; ++i) L[st_base + i * 16] = h2[i];
        __asm__ volatile("s_wait_dscnt 0x0" ::: "memory");
        v2f a3[2];
#pragma unroll
        for (int j = 0; j < 2; ++j) a3[j] = *(const v2f*)(L + ld_base + 4 * j);

        // ---- layer 3: out = h2 @ W3 + b3, K=8 via 2 chained WMMAs -----------
#pragma unroll
        for (int i = 0; i < 8; ++i) acc[i] = c3;
        v8f o = wmma_f32_4(a3[0], B3[0], acc);
        o = wmma_f32_4(a3[1], B3[1], o);

        // ---- store: lane holds out[base + 8*hi + i][nlo] for nlo < 3 --------
        if (nlo < 3) {
            const int r0 = base + 8 * hi;
#pragma unroll
            for (int i = 0; i < 8; ++i) {
                const int r = r0 + i;
                if (!GUARD || r < Brows)
                    __builtin_nontemporal_store(o[i], out + r * 3 + nlo);
            }
        }
    }
}

extern "C" void kernel_launch(void* const* d_in, const int* in_sizes, int n_in,
                              void* d_out, int out_size, void* d_ws, size_t ws_size,
                              hipStream_t stream) {
    const float* x  = (const float*)d_in[0];
    const float* W1 = (const float*)d_in[1];
    const float* b1 = (const float*)d_in[2];
    const float* W2 = (const float*)d_in[3];
    const float* b2 = (const float*)d_in[4];
    const float* W3 = (const float*)d_in[5];
    const float* b3 = (const float*)d_in[6];
    float* out = (float*)d_out;

    const int Brows = in_sizes[0] / 4;            // x is (B,4)
    const int tiles = (Brows + 15) / 16;          // 16 rows per wave-tile
    int blocks = (tiles + 7) / 8;                 // 8 waves per block
    if (blocks > 4096) blocks = 4096;             // grid-stride: ~4 tiles/wave
    if (blocks < 1) blocks = 1;

    if ((Brows & 15) == 0) {
        // exact multiple of 16 rows (B = 2,000,000): branch-free inner loop
        mlp3_wmma_kernel<false><<<blocks, 256, 0, stream>>>(
            x, W1, b1, W2, b2, W3, b3, out, Brows, tiles);
    } else {
        mlp3_wmma_kernel<true><<<blocks, 256, 0, stream>>>(
            x, W1, b1, W2, b2, W3, b3, out, Brows, tiles);
    }
}